// RY_30494267801972
// MI455X (gfx1250) — compile-verified
//
#include <hip/hip_runtime.h>
#include <math.h>

// RY(θ) on qubit 6 of a 13-qubit batch-256 state, xr & xi simultaneously.
//
// View x as (slab, a, j): slab = b*64+i (16384 slabs), a in {0,1} (stride
// 64 floats = 256B), j in [0,64).
//   out0 = c*x0 + s*x1 ; out1 = -s*x0 + c*x1   (c = cos θ, s = sin θ)
//
// Memory-bound: 33.5 MB total traffic -> ~1.44 us floor at 23.3 TB/s.
// Rotation expressed as V_WMMA_F32_16X16X4_F32:
//   A (16x4, ISA-documented layout) = coefficients:
//       row0=[ c, s,0,0] row1=[-s, c,0,0] row2=[0,0, c, s] row3=[0,0,-s, c]
//   B (4x16) = data: row0=xr_a0, row1=xr_a1, row2=xi_a0, row3=xi_a1,
//              columns = 16 consecutive j sites
//   D rows 0..3 = out_xr0/out_xr1/out_xi0/out_xi1, each contiguous across
//              lanes 0-15 (D VGPR r, lanes<16 -> D[r,n]); rows 4..15 zero.
// B lane layout assumption (not tabulated in ISA doc; transpose-analog of A):
//   VGPR0: lanes0-15 = B[0,n], lanes16-31 = B[1,n]
//   VGPR1: lanes0-15 = B[2,n], lanes16-31 = B[3,n]
// D row pairs are merged into full 32-lane b32 stores via ds_swizzle SWAPX16
// + v_cndmask, so EXEC stays all-1s (WMMA requirement) and every memory op
// is a full-wave 128B access (two contiguous 64B segments 256B apart).
//
// Each wave processes SLABS_PER_WAVE slabs to amortize the one-time precise
// cosf/sinf + A-operand setup and to deepen the per-wave load pipeline.

typedef __attribute__((ext_vector_type(2))) float v2f;
typedef __attribute__((ext_vector_type(8))) float v8f;

#define SWZ_SWAPX16 0x401F   // group-of-32: xor=0x10, or=0, and=0x1F
#define SLABS_PER_WAVE 2

__global__ __launch_bounds__(256) void ry_wmma_kernel(
    const float* __restrict__ xr, const float* __restrict__ xi,
    const float* __restrict__ params,
    float* __restrict__ outR, float* __restrict__ outI)
{
    const int lane = threadIdx.x & 31;
    const int wave = blockIdx.x * 8 + (threadIdx.x >> 5);

    const float th = params[0];
    const float c = cosf(th);
    const float s = sinf(th);

    // A operand (16x4 f32, ISA-documented layout):
    // lanes 0-15:  VGPR0 = A[l,0], VGPR1 = A[l,1]
    // lanes 16-31: VGPR0 = A[l,2], VGPR1 = A[l,3]   (l = lane & 15)
    // lo half -> coefficient rows 0,1 (xr channels); hi half -> rows 2,3 (xi).
    const int l = lane & 15;
    const int r = (lane < 16) ? l : (l - 2);   // r==0/r==1 pick the coeff rows
    v2f A;
    A.x = (r == 0) ? c : ((r == 1) ? -s : 0.0f);
    A.y = (r == 0) ? s : ((r == 1) ?  c : 0.0f);

    // Per-lane element offset inside a slab: lanes 0-15 -> row a=0 (j=lane),
    // lanes 16-31 -> row a=1 (j=lane-16), i.e. +64 floats.
    const int laneOff = lane + ((lane >> 4) * 48);

    #pragma unroll
    for (int q = 0; q < SLABS_PER_WAVE; ++q) {
        const int slab = wave * SLABS_PER_WAVE + q;
        const int base = slab * 128;             // 128 floats per slab

        #pragma unroll
        for (int j0 = 0; j0 < 64; j0 += 16) {
            const int e = base + j0 + laneOff;

            v2f B;
            B.x = __builtin_nontemporal_load(xr + e);  // B rows 0,1 (xr pair)
            B.y = __builtin_nontemporal_load(xi + e);  // B rows 2,3 (xi pair)

            v8f C = {};
            v8f D = __builtin_amdgcn_wmma_f32_16x16x4_f32(
                /*neg_a=*/false, A, /*neg_b=*/false, B,
                /*c_mod=*/(short)0, C, /*reuse_a=*/false, /*reuse_b=*/false);

            // Merge D row pairs into full-wave stores:
            //  R0: lanes0-15 = D[0,n] (out_xr a=0), lanes16-31 = D[1,n] (a=1)
            //  R1: lanes0-15 = D[2,n] (out_xi a=0), lanes16-31 = D[3,n] (a=1)
            const float d1s = __int_as_float(
                __builtin_amdgcn_ds_swizzle(__float_as_int(D[1]), SWZ_SWAPX16));
            const float d3s = __int_as_float(
                __builtin_amdgcn_ds_swizzle(__float_as_int(D[3]), SWZ_SWAPX16));
            const float R0 = (lane < 16) ? D[0] : d1s;
            const float R1 = (lane < 16) ? D[2] : d3s;

            __builtin_nontemporal_store(R0, outR + e);
            __builtin_nontemporal_store(R1, outI + e);
        }
    }
}

extern "C" void kernel_launch(void* const* d_in, const int* in_sizes, int n_in,
                              void* d_out, int out_size, void* d_ws, size_t ws_size,
                              hipStream_t stream) {
    const float* xr     = (const float*)d_in[0];   // (256, 8192) f32
    const float* xi     = (const float*)d_in[1];   // (256, 8192) f32
    const float* params = (const float*)d_in[2];   // (1,) f32

    float* outR = (float*)d_out;                   // rotated xr
    float* outI = outR + (size_t)in_sizes[0];      // rotated xi (tuple concat)

    const int nslab  = in_sizes[0] >> 7;           // elements / 128 = 16384
    const int waves  = nslab / SLABS_PER_WAVE;     // 8192
    const int blocks = waves >> 3;                 // 8 waves per block -> 1024

    hipLaunchKernelGGL(ry_wmma_kernel, dim3(blocks), dim3(256), 0, stream,
                       xr, xi, params, outR, outI);
}